// DynAngioTheoreticalIntGammaAllRFAnalytic_88974542504565
// MI455X (gfx1250) — compile-verified
//
#include <hip/hip_runtime.h>
#include <cstdint>

#define NT   144
#define TAUc 1.8f
#define T1B  1.65f

typedef uint32_t v4u __attribute__((ext_vector_type(4)));
typedef int      v4i __attribute__((ext_vector_type(4)));
typedef int      v8i __attribute__((ext_vector_type(8)));

__device__ __forceinline__ float frcp(float x) { return __builtin_amdgcn_rcpf(x); }

// Gamma(1+u), 0 <= u < 1  (A&S 6.1.36, |err| <= 3e-7)
__device__ __forceinline__ float gamma1pu(float u) {
    float g = 0.035868343f;
    g = fmaf(g, u, -0.193527818f);
    g = fmaf(g, u,  0.482199394f);
    g = fmaf(g, u, -0.756704078f);
    g = fmaf(g, u,  0.918206857f);
    g = fmaf(g, u, -0.897056937f);
    g = fmaf(g, u,  0.988205891f);
    g = fmaf(g, u, -0.577191652f);
    g = fmaf(g, u,  1.0f);
    return g;
}

// Unnormalized P*Gamma(a): E(x) * (1/a) * S, with S the nested (Horner) form of
// the N-term Kummer series; rn[n] = 1/(a+n) precomputed per row.
// 2 VALU/term: one off-chain mul (x*rn) + one fma.  GUARD handles x <= 0.
template<int N, bool GUARD>
__device__ __forceinline__ float igamp_u(float x, float a, const float* rn) {
    const float xc = GUARD ? fmaxf(x, 1e-30f) : x;
    const float e  = __expf(fmaf(a, __logf(xc), -xc));   // x^a * e^-x
    float S = 1.0f;
#pragma unroll
    for (int n = N; n >= 1; --n)
        S = fmaf(xc * rn[n], S, 1.0f);
    float P = e * rn[0] * S;                             // missing 1/Gamma(a): folded into pref
    if (GUARD) P = (x > 0.0f) ? P : 0.0f;
    return P;
}

// One-time column table: w[j] = R[j] * sin(deg2rad(alpha[j]))
__global__ void dynangio_prep(const float* __restrict__ alpha,
                              const float* __restrict__ R,
                              float* __restrict__ w) {
    const int j = threadIdx.x;
    if (j < NT) w[j] = R[j] * __sinf(alpha[j] * 0.017453292519943295f);
}

__global__ __launch_bounds__(256) void dynangio_main(
        const float4* __restrict__ x,
        const float*  __restrict__ t,
        const float*  __restrict__ w,
        float* __restrict__ out, int B) {
    __shared__ float sm_t[NT];
    __shared__ float sm_w[NT];
    const int tid = threadIdx.x;

    // ---- Stage t[] via Tensor Data Mover (wave 0 only; EXEC ignored by TDM) ----
    if (tid < 32) {
        const uint32_t lds_t = (uint32_t)(uintptr_t)&sm_t[0];
        const uint64_t ga    = (uint64_t)(uintptr_t)t;
        v4u g0;
        g0[0] = 1u;                                                 // count=1, user descriptor
        g0[1] = lds_t;                                              // lds_addr
        g0[2] = (uint32_t)ga;                                       // global_addr[31:0]
        g0[3] = (uint32_t)((ga >> 32) & 0x01FFFFFFu) | 0x80000000u; // addr[56:32] | type=2
        v8i g1;
        g1[0] = (int)(2u << 16);     // data_size = 4 bytes
        g1[1] = (int)(NT << 16);     // tensor_dim0 = 144
        g1[2] = (int)(1u << 16);     // tensor_dim1 = 1
        g1[3] = (int)(NT << 16);     // tile_dim0   = 144
        g1[4] = 1;                   // tile_dim1   = 1
        g1[5] = NT;                  // tensor_dim0_stride = 144
        g1[6] = 0;
        g1[7] = 0;
        v4i g2 = {0, 0, 0, 0};
        v4i g3 = {0, 0, 0, 0};
#if defined(__clang_major__) && __clang_major__ >= 23
        v8i g4 = {0, 0, 0, 0, 0, 0, 0, 0};
        __builtin_amdgcn_tensor_load_to_lds(g0, g1, g2, g3, g4, 0);
#else
        __builtin_amdgcn_tensor_load_to_lds(g0, g1, g2, g3, 0);
#endif
        __builtin_amdgcn_s_wait_tensorcnt(0);
    }

    // ---- Stage w[] via async global->LDS loads (ASYNCcnt path) ----
    if (tid < NT) {
        const uint32_t lds_w = (uint32_t)(uintptr_t)&sm_w[tid];
        const float*   gp    = w + tid;
        asm volatile("global_load_async_to_lds_b32 %0, %1, off"
                     :: "v"(lds_w), "v"(gp) : "memory");
        asm volatile("s_wait_asynccnt 0" ::: "memory");
    }
    __syncthreads();

    // ---- One wave per row; lanes sweep the 144 columns (coalesced stores) ----
    const int wave = tid >> 5;
    const int lane = tid & 31;
    const int row  = blockIdx.x * 8 + wave;
    if (row >= B) return;

    const float4 xr = x[row];                  // broadcast b128 load
    const float delta_t = xr.x, s = xr.y, p = xr.z, amp = xr.w;

    const float u      = p * s;
    const float a      = 1.0f + u;             // a in [1,2)
    const float sprime = s + (1.0f / T1B);
    const float rgam   = frcp(gamma1pu(u));    // 1/Gamma(a)
    const float pref   = amp * 2.0f * __expf(-delta_t * (1.0f / T1B))
                             * __powf(s * frcp(sprime), a)
                             * rgam;           // 1/Gamma(a) folded in once per row
    const float sptau  = sprime * TAUc;

    // Per-row reciprocal table: rn[n] = 1/(a+n). 25 rcps amortized over 144 cols.
    float rn[25];
#pragma unroll
    for (int n = 0; n < 25; ++n) rn[n] = frcp(a + (float)n);

    float* __restrict__ orow = out + (size_t)row * NT;
    for (int it = 0; it < 5; ++it) {
        const int j = lane + it * 32;
        if (j < NT) {
            const float td = sm_t[j] - delta_t;
            const float x1 = sprime * td;            // always > 0, <= ~6.3
            const float x2 = x1 - sptau;             // == sprime*(td - TAU), may be <= 0
            const float P1 = igamp_u<24, false>(x1, a, rn);
            const float P2 = igamp_u<16, true >(x2, a, rn);
            const float v  = pref * sm_w[j] * (P1 - P2);
            __builtin_nontemporal_store(v, &orow[j]);   // streamed output
        }
    }
}

extern "C" void kernel_launch(void* const* d_in, const int* in_sizes, int n_in,
                              void* d_out, int out_size, void* d_ws, size_t ws_size,
                              hipStream_t stream) {
    const float* x     = (const float*)d_in[0];   // [B,4]
    const float* t     = (const float*)d_in[1];   // [144]
    const float* alpha = (const float*)d_in[2];   // [144]
    const float* R     = (const float*)d_in[3];   // [144]
    float*       out   = (float*)d_out;           // [B,144]
    float*       w     = (float*)d_ws;            // 144 floats scratch

    const int B = in_sizes[0] / 4;

    dynangio_prep<<<1, 256, 0, stream>>>(alpha, R, w);

    const int blocks = (B + 7) / 8;               // 8 rows (waves) per 256-thread block
    dynangio_main<<<blocks, 256, 0, stream>>>((const float4*)x, t, w, out, B);
}